// CellRetrievalNetwork_41927470744123
// MI455X (gfx1250) — compile-verified
//
#include <hip/hip_runtime.h>
#include <hip/hip_bf16.h>
#include <math.h>

typedef __attribute__((ext_vector_type(2))) float v2f;
typedef __attribute__((ext_vector_type(8))) float v8f;

constexpr int kE   = 256;   // embed dim
constexpr int kK   = 8;     // knn neighbors
constexpr int kB   = 64;    // cells
constexpr int kP   = 512;   // points per cell
constexpr int kN   = kB * kP;
constexpr int kHID = 64;

__device__ inline v8f wmma_f32(v2f a, v2f b, v8f c) {
  // D = A(16x4,f32) * B(4x16,f32) + C(16x16,f32)
  return __builtin_amdgcn_wmma_f32_16x16x4_f32(
      /*neg_a=*/false, a, /*neg_b=*/false, b,
      /*c_mod=*/(short)0, c, /*reuse_a=*/false, /*reuse_b=*/false);
}

__device__ inline float wave_sum(float v) {
#pragma unroll
  for (int off = 16; off > 0; off >>= 1) v += __shfl_xor(v, off, 32);
  return v;
}

// ---------------------------------------------------------------------------
// Kernel 0: weight transpose  WT[n][k] = W[k][n]  (coalesced writes).
// Makes every WMMA B-fragment a single contiguous 8B load.
// ---------------------------------------------------------------------------
__global__ __launch_bounds__(256) void k_transpose(const float* __restrict__ W,
                                                   float* __restrict__ WT,
                                                   int Kdim, int Ndim) {
  const int t = blockIdx.x * 256 + threadIdx.x;
  if (t >= Kdim * Ndim) return;
  const int n = t / Kdim;
  const int k = t - n * Kdim;
  WT[t] = W[(size_t)k * Ndim + n];
}

// ---------------------------------------------------------------------------
// Kernel 1: per-object embeddings (class gather + 2 small MLPs, each L2
// normalized) staged into LDS as cat[16][768], then merge GEMM (768->256)
// via WMMA. One wave per 16 objects.
// ---------------------------------------------------------------------------
__global__ __launch_bounds__(32) void k_embed_merge(
    const float* __restrict__ class_table,
    const float* __restrict__ cW1, const float* __restrict__ cb1,
    const float* __restrict__ cW2, const float* __restrict__ cb2,
    const float* __restrict__ pW1, const float* __restrict__ pb1,
    const float* __restrict__ pW2, const float* __restrict__ pb2,
    const float* __restrict__ mergeWT, const float* __restrict__ mergeb,
    const float* __restrict__ positions, const float* __restrict__ colors,
    const int* __restrict__ cls,
    float* __restrict__ emb) {
  __shared__ float cat[16][772];   // 768 + pad (row stride mult of 4)
  __shared__ float hid[kHID];
  const int lane = threadIdx.x;
  const int g0 = blockIdx.x * 16;

  // --- class embedding: gather + normalize -> cat[o][0:256]
  for (int o = 0; o < 16; ++o) {
    const int idx = cls[g0 + o];
    float v[8];
    float ssq = 0.0f;
#pragma unroll
    for (int i = 0; i < 8; ++i) {
      v[i] = class_table[(size_t)idx * kE + lane + i * 32];
      ssq += v[i] * v[i];
    }
    ssq = wave_sum(ssq);
    const float s = 1.0f / fmaxf(sqrtf(ssq), 1e-12f);
#pragma unroll
    for (int i = 0; i < 8; ++i) cat[o][lane + i * 32] = v[i] * s;
  }

  // --- color MLP (which=0) -> cat[o][256:512]; pos MLP (which=1) -> [512:768]
  for (int which = 0; which < 2; ++which) {
    const float* W1 = which ? pW1 : cW1;
    const float* B1 = which ? pb1 : cb1;
    const float* W2 = which ? pW2 : cW2;
    const float* B2 = which ? pb2 : cb2;
    const float* feat = which ? positions : colors;
    const int off = which ? 512 : 256;
    for (int o = 0; o < 16; ++o) {
      const float f0 = feat[(size_t)(g0 + o) * 3 + 0];
      const float f1 = feat[(size_t)(g0 + o) * 3 + 1];
      const float f2 = feat[(size_t)(g0 + o) * 3 + 2];
      for (int j = lane; j < kHID; j += 32) {
        float h = fmaf(f0, W1[0 * kHID + j],
                  fmaf(f1, W1[1 * kHID + j],
                  fmaf(f2, W1[2 * kHID + j], B1[j])));
        hid[j] = fmaxf(h, 0.0f);
      }
      __syncthreads();
      float v[8];
      float ssq = 0.0f;
#pragma unroll
      for (int i = 0; i < 8; ++i) {
        const int e = lane + i * 32;
        float acc = B2[e];
        for (int j = 0; j < kHID; ++j) acc = fmaf(hid[j], W2[j * kE + e], acc);
        v[i] = acc;
        ssq += acc * acc;
      }
      ssq = wave_sum(ssq);
      const float s = 1.0f / fmaxf(sqrtf(ssq), 1e-12f);
#pragma unroll
      for (int i = 0; i < 8; ++i) cat[o][off + lane + i * 32] = v[i] * s;
      __syncthreads();
    }
  }
  __syncthreads();

  // --- merge GEMM: emb[16 x 256] = cat[16 x 768] @ mergeW[768 x 256] + b
  const int Mi = lane & 15;
  const int Kb = (lane >> 4) * 2;
  const int rowHalf = (lane >> 4) * 8;
  for (int nt = 0; nt < 16; ++nt) {
    const float* wrow = mergeWT + (size_t)(nt * 16 + Mi) * 768;
    if (nt + 1 < 16)
      __builtin_prefetch(mergeWT + (size_t)((nt + 1) * 16 + Mi) * 768, 0, 1);
    v8f acc = {};
    for (int k = 0; k < 768; k += 4) {
      const v2f a = *(const v2f*)&cat[Mi][k + Kb];
      const v2f bb = *(const v2f*)&wrow[k + Kb];
      acc = wmma_f32(a, bb, acc);
    }
    const int Nc = nt * 16 + Mi;
    const float bias = mergeb[Nc];
#pragma unroll
    for (int r = 0; r < 8; ++r)
      emb[(size_t)(g0 + r + rowHalf) * kE + Nc] = acc[r] + bias;
  }
}

// ---------------------------------------------------------------------------
// Kernel 2: per-row squared norms of emb
// ---------------------------------------------------------------------------
__global__ __launch_bounds__(256) void k_sqnorm(const float* __restrict__ emb,
                                                float* __restrict__ sqn) {
  const int wave = threadIdx.x >> 5;
  const int lane = threadIdx.x & 31;
  const int row = blockIdx.x * 8 + wave;
  float s = 0.0f;
#pragma unroll
  for (int i = 0; i < 8; ++i) {
    const float v = emb[(size_t)row * kE + lane + i * 32];
    s += v * v;
  }
  s = wave_sum(s);
  if (lane == 0) sqn[row] = s;
}

// ---------------------------------------------------------------------------
// Kernel 3: per-cell Gram tile (16 rows x 512 cols) via WMMA, then top-8
// (smallest score = sq[n] - 2<x_m,x_n>; same ranking as reference dist,
// and self is provably the minimum). One wave per (cell, 16-row tile).
// B-side fragments (emb^T) are contiguous 8B loads.
// ---------------------------------------------------------------------------
__global__ __launch_bounds__(32) void k_knn(const float* __restrict__ emb,
                                            const float* __restrict__ sqn,
                                            int* __restrict__ knn) {
  __shared__ float arow[16][260];   // A rows staged
  __shared__ float score[16][516];  // 16 rows x 512 cols (+pad)
  const int b = blockIdx.x >> 5;    // 32 row tiles per cell
  const int rt = blockIdx.x & 31;
  const int lane = threadIdx.x;
  const int base = b * kP;
  const int pr = rt * 16;

  for (int r = 0; r < 16; ++r)
    for (int i = lane; i < kE; i += 32)
      arow[r][i] = emb[(size_t)(base + pr + r) * kE + i];
  __syncthreads();

  const int Mi = lane & 15;
  const int Kb = (lane >> 4) * 2;
  const int rowHalf = (lane >> 4) * 8;
  for (int ct = 0; ct < 32; ++ct) {
    const float* brow = emb + (size_t)(base + ct * 16 + Mi) * kE;
    if (ct + 1 < 32)
      __builtin_prefetch(emb + (size_t)(base + (ct + 1) * 16 + Mi) * kE, 0, 1);
    v8f acc = {};
    for (int k = 0; k < kE; k += 4) {
      const v2f a = *(const v2f*)&arow[Mi][k + Kb];
      const v2f bb = *(const v2f*)&brow[k + Kb];
      acc = wmma_f32(a, bb, acc);
    }
    const int Nc = ct * 16 + Mi;
    const float sn = sqn[base + Nc];
#pragma unroll
    for (int r = 0; r < 8; ++r)
      score[r + rowHalf][Nc] = sn - 2.0f * acc[r];
  }
  __syncthreads();

  // top-8 per row; 2 lanes cooperate per row
  const int row = lane >> 1;
  const int half = lane & 1;
  float bv[8];
  int bi[8];
#pragma unroll
  for (int j = 0; j < 8; ++j) { bv[j] = 3.4e38f; bi[j] = 0x7fffffff; }
  for (int c = half; c < kP; c += 2) {
    const float v = score[row][c];
    if (v < bv[7] || (v == bv[7] && c < bi[7])) {
      int j = 7;
      while (j > 0 && (v < bv[j - 1] || (v == bv[j - 1] && c < bi[j - 1]))) {
        bv[j] = bv[j - 1];
        bi[j] = bi[j - 1];
        --j;
      }
      bv[j] = v;
      bi[j] = c;
    }
  }
  float ov[8];
  int oi[8];
#pragma unroll
  for (int j = 0; j < 8; ++j) {
    ov[j] = __shfl_xor(bv[j], 1, 32);
    oi[j] = __shfl_xor(bi[j], 1, 32);
  }
  int mi_[8];
  int ia = 0, ib = 0;
#pragma unroll
  for (int j = 0; j < 8; ++j) {
    const bool takeA = (ib >= 8) ||
        (ia < 8 && (bv[ia] < ov[ib] || (bv[ia] == ov[ib] && bi[ia] < oi[ib])));
    if (takeA) { mi_[j] = bi[ia]; ++ia; }
    else       { mi_[j] = oi[ib]; ++ib; }
  }
  if (half == 0) {
#pragma unroll
    for (int j = 0; j < 8; ++j)
      knn[(size_t)(base + pr + row) * kK + j] = mi_[j];
  }
}

// ---------------------------------------------------------------------------
// Kernel 4: EdgeConv. One wave handles 2 points x 8 neighbors = 16 msg rows.
// msg[16][512] -> WMMA GEMM1 (+bias,relu,bn) -> h[16][256] -> WMMA GEMM2
// (+bias) -> max over k (free from D-fragment layout) -> x1.
// Weights consumed transposed: one 8B load per B fragment, sequential walk.
// ---------------------------------------------------------------------------
__global__ __launch_bounds__(32) void k_edgeconv(
    const float* __restrict__ emb, const int* __restrict__ knn,
    const float* __restrict__ gW1T, const float* __restrict__ gb1,
    const float* __restrict__ gga, const float* __restrict__ gbe,
    const float* __restrict__ gW2T, const float* __restrict__ gb2,
    float* __restrict__ x1) {
  __shared__ float msg[16][516];
  __shared__ float h[16][260];
  const int b = blockIdx.x >> 8;   // 256 blocks per cell (P/2)
  const int p0 = (blockIdx.x & 255) * 2;
  const int lane = threadIdx.x;
  const int base = b * kP;

  for (int r = 0; r < 16; ++r) {
    const int pi = p0 + (r >> 3);
    const int nk = r & 7;
    const int j = knn[(size_t)(base + pi) * kK + nk];
    for (int e = lane; e < kE; e += 32) {
      const float xi = emb[(size_t)(base + pi) * kE + e];
      const float xj = emb[(size_t)(base + j) * kE + e];
      msg[r][e] = xi;
      msg[r][kE + e] = xj - xi;
    }
  }
  __syncthreads();

  const int Mi = lane & 15;
  const int Kb = (lane >> 4) * 2;
  const int rowHalf = (lane >> 4) * 8;

  for (int nt = 0; nt < 16; ++nt) {
    const float* wrow = gW1T + (size_t)(nt * 16 + Mi) * 512;
    if (nt + 1 < 16)
      __builtin_prefetch(gW1T + (size_t)((nt + 1) * 16 + Mi) * 512, 0, 1);
    v8f acc = {};
    for (int k = 0; k < 2 * kE; k += 4) {
      const v2f a = *(const v2f*)&msg[Mi][k + Kb];
      const v2f bb = *(const v2f*)&wrow[k + Kb];
      acc = wmma_f32(a, bb, acc);
    }
    const int Nc = nt * 16 + Mi;
    const float b1 = gb1[Nc], ga = gga[Nc], be = gbe[Nc];
#pragma unroll
    for (int r = 0; r < 8; ++r) {
      const float v = fmaxf(acc[r] + b1, 0.0f);
      h[r + rowHalf][Nc] = v * ga + be;
    }
  }
  __syncthreads();

  for (int nt = 0; nt < 16; ++nt) {
    const float* wrow = gW2T + (size_t)(nt * 16 + Mi) * kE;
    if (nt + 1 < 16)
      __builtin_prefetch(gW2T + (size_t)((nt + 1) * 16 + Mi) * kE, 0, 1);
    v8f acc = {};
    for (int k = 0; k < kE; k += 4) {
      const v2f a = *(const v2f*)&h[Mi][k + Kb];
      const v2f bb = *(const v2f*)&wrow[k + Kb];
      acc = wmma_f32(a, bb, acc);
    }
    const int Nc = nt * 16 + Mi;
    // rows 0..7 (this lane half) are the 8 neighbors of one point
    float m = acc[0];
#pragma unroll
    for (int r = 1; r < 8; ++r) m = fmaxf(m, acc[r]);
    m += gb2[Nc];
    const int p = p0 + (lane >> 4);
    x1[(size_t)(base + p) * kE + Nc] = m;
  }
}

// ---------------------------------------------------------------------------
// Kernel 5: global max pool over P per cell
// ---------------------------------------------------------------------------
__global__ __launch_bounds__(256) void k_pool(const float* __restrict__ x1,
                                              float* __restrict__ pooled) {
  const int b = blockIdx.x;
  const int e = threadIdx.x;
  float m = -3.4e38f;
  for (int p = 0; p < kP; ++p)
    m = fmaxf(m, x1[(size_t)(b * kP + p) * kE + e]);
  pooled[b * kE + e] = m;
}

// ---------------------------------------------------------------------------
// Kernel 6: final MLP (WMMA) + row L2 normalize. One wave per 16 cells.
// ---------------------------------------------------------------------------
__global__ __launch_bounds__(32) void k_final(
    const float* __restrict__ pooled,
    const float* __restrict__ lW1T, const float* __restrict__ lb1,
    const float* __restrict__ lW2T, const float* __restrict__ lb2,
    float* __restrict__ out) {
  __shared__ float A[16][260];
  __shared__ float H[16][260];
  __shared__ float O[16][260];
  const int lane = threadIdx.x;
  const int r0 = blockIdx.x * 16;

  for (int r = 0; r < 16; ++r)
    for (int e = lane; e < kE; e += 32)
      A[r][e] = pooled[(size_t)(r0 + r) * kE + e];
  __syncthreads();

  const int Mi = lane & 15;
  const int Kb = (lane >> 4) * 2;
  const int rowHalf = (lane >> 4) * 8;

  for (int nt = 0; nt < 16; ++nt) {
    const float* wrow = lW1T + (size_t)(nt * 16 + Mi) * kE;
    v8f acc = {};
    for (int k = 0; k < kE; k += 4) {
      const v2f a = *(const v2f*)&A[Mi][k + Kb];
      const v2f bb = *(const v2f*)&wrow[k + Kb];
      acc = wmma_f32(a, bb, acc);
    }
    const int Nc = nt * 16 + Mi;
    const float b1 = lb1[Nc];
#pragma unroll
    for (int r = 0; r < 8; ++r) H[r + rowHalf][Nc] = fmaxf(acc[r] + b1, 0.0f);
  }
  __syncthreads();

  for (int nt = 0; nt < 16; ++nt) {
    const float* wrow = lW2T + (size_t)(nt * 16 + Mi) * kE;
    v8f acc = {};
    for (int k = 0; k < kE; k += 4) {
      const v2f a = *(const v2f*)&H[Mi][k + Kb];
      const v2f bb = *(const v2f*)&wrow[k + Kb];
      acc = wmma_f32(a, bb, acc);
    }
    const int Nc = nt * 16 + Mi;
    const float b2 = lb2[Nc];
#pragma unroll
    for (int r = 0; r < 8; ++r) O[r + rowHalf][Nc] = acc[r] + b2;
  }
  __syncthreads();

  const int row = lane >> 1;
  const int half = lane & 1;
  float s = 0.0f;
  for (int e = half * 128; e < half * 128 + 128; ++e) {
    const float v = O[row][e];
    s += v * v;
  }
  s += __shfl_xor(s, 1, 32);
  const float sc = 1.0f / fmaxf(sqrtf(s), 1e-12f);
  for (int e = half * 128; e < half * 128 + 128; ++e)
    out[(size_t)(r0 + row) * kE + e] = O[row][e] * sc;
}

// ---------------------------------------------------------------------------
extern "C" void kernel_launch(void* const* d_in, const int* in_sizes, int n_in,
                              void* d_out, int out_size, void* d_ws,
                              size_t ws_size, hipStream_t stream) {
  const float* class_table = (const float*)d_in[0];
  const float* colorW1 = (const float*)d_in[1];
  const float* colorb1 = (const float*)d_in[2];
  const float* colorW2 = (const float*)d_in[3];
  const float* colorb2 = (const float*)d_in[4];
  const float* posW1 = (const float*)d_in[5];
  const float* posb1 = (const float*)d_in[6];
  const float* posW2 = (const float*)d_in[7];
  const float* posb2 = (const float*)d_in[8];
  const float* mergeW = (const float*)d_in[9];
  const float* mergeb = (const float*)d_in[10];
  const float* gW1 = (const float*)d_in[11];
  const float* gb1 = (const float*)d_in[12];
  const float* g_gamma = (const float*)d_in[13];
  const float* g_beta = (const float*)d_in[14];
  const float* gW2 = (const float*)d_in[15];
  const float* gb2 = (const float*)d_in[16];
  const float* lW1 = (const float*)d_in[17];
  const float* lb1 = (const float*)d_in[18];
  const float* lW2 = (const float*)d_in[19];
  const float* lb2 = (const float*)d_in[20];
  const float* positions = (const float*)d_in[21];
  const float* colors = (const float*)d_in[22];
  const int* class_indices = (const int*)d_in[23];
  (void)in_sizes; (void)n_in; (void)out_size; (void)ws_size;

  // workspace carve-up (~70.5 MB)
  float* emb = (float*)d_ws;                       // N*E
  float* sqn = emb + (size_t)kN * kE;              // N
  float* x1 = sqn + kN;                            // N*E
  float* pooled = x1 + (size_t)kN * kE;            // B*E
  float* mergeWT = pooled + (size_t)kB * kE;       // 256*768
  float* gW1T = mergeWT + 256 * 768;               // 256*512
  float* gW2T = gW1T + 256 * 512;                  // 256*256
  float* lW1T = gW2T + 256 * 256;                  // 256*256
  float* lW2T = lW1T + 256 * 256;                  // 256*256
  int* knn = (int*)(lW2T + 256 * 256);             // B*P*K ints

  // weight transposes (coalesced writes)
  k_transpose<<<(768 * 256 + 255) / 256, 256, 0, stream>>>(mergeW, mergeWT,
                                                           768, 256);
  k_transpose<<<(512 * 256 + 255) / 256, 256, 0, stream>>>(gW1, gW1T, 512,
                                                           256);
  k_transpose<<<(256 * 256 + 255) / 256, 256, 0, stream>>>(gW2, gW2T, 256,
                                                           256);
  k_transpose<<<(256 * 256 + 255) / 256, 256, 0, stream>>>(lW1, lW1T, 256,
                                                           256);
  k_transpose<<<(256 * 256 + 255) / 256, 256, 0, stream>>>(lW2, lW2T, 256,
                                                           256);

  k_embed_merge<<<kN / 16, 32, 0, stream>>>(
      class_table, colorW1, colorb1, colorW2, colorb2, posW1, posb1, posW2,
      posb2, mergeWT, mergeb, positions, colors, class_indices, emb);

  k_sqnorm<<<kN / 8, 256, 0, stream>>>(emb, sqn);

  k_knn<<<kB * (kP / 16), 32, 0, stream>>>(emb, sqn, knn);

  k_edgeconv<<<kB * (kP / 2), 32, 0, stream>>>(emb, knn, gW1T, gb1, g_gamma,
                                               g_beta, gW2T, gb2, x1);

  k_pool<<<kB, 256, 0, stream>>>(x1, pooled);

  k_final<<<kB / 16, 32, 0, stream>>>(pooled, lW1T, lb1, lW2T, lb2,
                                      (float*)d_out);
}